// BrickPlacementGNN_31430570672234
// MI455X (gfx1250) — compile-verified
//
#include <hip/hip_runtime.h>
#include <hip/hip_bf16.h>
#include <math.h>

typedef __attribute__((ext_vector_type(2))) float v2f;
typedef __attribute__((ext_vector_type(8))) float v8f;

constexpr int NN   = 50000;
constexpr int EE   = 1000000;
constexpr int EP   = EE + NN;     // edges + self loops
constexpr int HID  = 256;         // H*C
constexpr int HEADS= 4;
constexpr int CH   = 64;
constexpr int EDIM = 16;
constexpr int NDIM = 32;
constexpr int ODIM = 128;
constexpr float NEG_SLOPE = 0.2f;

// ---------------- workspace layout (in floats) ----------------
constexpr size_t OFF_B0    = 0;                          // [NN*HID] linear outputs
constexpr size_t OFF_B1    = OFF_B0 + (size_t)NN*HID;    // [NN*HID] aggregation / post-act
constexpr size_t OFF_LG    = OFF_B1 + (size_t)NN*HID;    // [EP*HEADS] logits
constexpr size_t OFF_EAL   = OFF_LG + (size_t)EP*HEADS;  // [NN*EDIM] self-loop edge attr
constexpr size_t OFF_DEG   = OFF_EAL + (size_t)NN*EDIM;  // [NN]
constexpr size_t OFF_M     = OFF_DEG + (size_t)NN;       // [NN*HEADS] segment max
constexpr size_t OFF_DEN   = OFF_M   + (size_t)NN*HEADS; // [NN*HEADS] segment sum
constexpr size_t OFF_ASRC  = OFF_DEN + (size_t)NN*HEADS; // [NN*HEADS]
constexpr size_t OFF_ADST  = OFF_ASRC+ (size_t)NN*HEADS; // [NN*HEADS]
constexpr size_t OFF_VE1   = OFF_ADST+ (size_t)NN*HEADS; // [HEADS*EDIM]
constexpr size_t OFF_VE2   = OFF_VE1 + (size_t)HEADS*EDIM;

// ---------------- utility kernels ----------------
__global__ void fill_kernel(float* p, float v, int n) {
  int i = blockIdx.x * blockDim.x + threadIdx.x;
  if (i < n) p[i] = v;
}

__global__ void deg_ea_kernel(const int* __restrict__ dst, const float* __restrict__ ea,
                              float* __restrict__ deg, float* __restrict__ ealoop) {
  int e = blockIdx.x * blockDim.x + threadIdx.x;
  if (e >= EE) return;
  int d = dst[e];
  atomicAdd(&deg[d], 1.0f);
  const float* p = ea + (size_t)e * EDIM;
  float* q = ealoop + (size_t)d * EDIM;
#pragma unroll
  for (int k = 0; k < EDIM; k++) atomicAdd(&q[k], p[k]);
}

__global__ void ea_div_kernel(float* __restrict__ ealoop, const float* __restrict__ deg) {
  int i = blockIdx.x * blockDim.x + threadIdx.x;
  if (i >= NN * EDIM) return;
  int n = i / EDIM;
  ealoop[i] /= fmaxf(deg[n], 1.0f);
}

// v_e[h*16+d] = sum_c We[d][h*64+c] * a_edge[h][c]   (64 threads)
__global__ void ve_kernel(const float* __restrict__ We, const float* __restrict__ aedge,
                          float* __restrict__ ve) {
  int idx = threadIdx.x;
  int h = idx >> 4, d = idx & 15;
  float s = 0.f;
#pragma unroll
  for (int c = 0; c < CH; c++) s += We[(size_t)d * HID + h * CH + c] * aedge[h * CH + c];
  ve[idx] = s;
}

// ---------------- WMMA fp32 GEMM: C[M,Ncol] = A[M,K] @ B[K,Ncol] (+ bias) ----------------
// Block = 8 waves -> 128 rows x 64 cols. B panel (K x 64) staged once in LDS.
// Each wave: 16x64 tile = 4 accumulators; per k-step one A b64 load + 4 WMMAs.
__global__ void gemm_wmma_kernel(const float* __restrict__ A, const float* __restrict__ B,
                                 float* __restrict__ C, const float* __restrict__ bias,
                                 int K, int Ncol, int rowTiles, int nPanels) {
  extern __shared__ float Bs[];                 // K * 64 floats
  int bn = blockIdx.x % nPanels;                // 64-col panel index
  int bm = blockIdx.x / nPanels;                // group of 8 row tiles
  int tid = threadIdx.x;
  int colBase = bn * 64;
  for (int e = tid; e < K * 64; e += 256) {     // cooperative B-panel load
    int kr = e >> 6, c = e & 63;
    Bs[e] = B[(size_t)kr * Ncol + colBase + c];
  }
  __syncthreads();
  int wave = tid >> 5;
  int rt = bm * 8 + wave;                       // row tile
  if (rt >= rowTiles) return;                   // wave-uniform; EXEC stays all-1s
  int lane = tid & 31, half = lane >> 4, l = lane & 15;
  const float* Arow = A + (size_t)(rt * 16 + l) * K;
  v8f acc[4] = {{0.f,0.f,0.f,0.f,0.f,0.f,0.f,0.f},
                {0.f,0.f,0.f,0.f,0.f,0.f,0.f,0.f},
                {0.f,0.f,0.f,0.f,0.f,0.f,0.f,0.f},
                {0.f,0.f,0.f,0.f,0.f,0.f,0.f,0.f}};
  for (int k = 0; k < K; k += 4) {
    int ka = k + half * 2;                      // lanes 16-31 hold K+2,K+3
    v2f a = *(const v2f*)(Arow + ka);           // 8B-aligned (K, ka even)
    const float* bp = Bs + ka * 64 + l;
    v2f b0, b1, b2, b3;
    b0.x = bp[0];  b0.y = bp[64];
    b1.x = bp[16]; b1.y = bp[80];
    b2.x = bp[32]; b2.y = bp[96];
    b3.x = bp[48]; b3.y = bp[112];
    acc[0] = __builtin_amdgcn_wmma_f32_16x16x4_f32(false, a, false, b0, (short)0, acc[0], false, false);
    acc[1] = __builtin_amdgcn_wmma_f32_16x16x4_f32(false, a, false, b1, (short)0, acc[1], false, false);
    acc[2] = __builtin_amdgcn_wmma_f32_16x16x4_f32(false, a, false, b2, (short)0, acc[2], false, false);
    acc[3] = __builtin_amdgcn_wmma_f32_16x16x4_f32(false, a, false, b3, (short)0, acc[3], false, false);
  }
#pragma unroll
  for (int ct = 0; ct < 4; ct++) {
    int col = colBase + ct * 16 + l;
    float bv = bias ? bias[col] : 0.f;
#pragma unroll
    for (int r = 0; r < 8; r++) {               // VGPR r -> row r (+8 for upper half)
      int row = rt * 16 + half * 8 + r;
      C[(size_t)row * Ncol + col] = acc[ct][r] + bv;
    }
  }
}

// per (node, head) attention coefficients from linear output
__global__ void alpha_kernel(const float* __restrict__ hlin, const float* __restrict__ aS,
                             const float* __restrict__ aD, float* __restrict__ asrc,
                             float* __restrict__ adst) {
  int idx = blockIdx.x * blockDim.x + threadIdx.x;
  if (idx >= NN * HEADS) return;
  int n = idx >> 2, h = idx & 3;
  const float* p = hlin + (size_t)n * HID + h * CH;
  float s = 0.f, d = 0.f;
#pragma unroll 8
  for (int c = 0; c < CH; c++) { float v = p[c]; s += v * aS[h * CH + c]; d += v * aD[h * CH + c]; }
  asrc[idx] = s;
  adst[idx] = d;
}

__device__ __forceinline__ void atomicMaxF(float* addr, float val) {
  if (val >= 0.f) atomicMax((int*)addr, __float_as_int(val));
  else            atomicMin((unsigned int*)addr, __float_as_uint(val));
}

// edge pass 1: logits (leaky-relu) + segment max over dst
__global__ void edge_logit_kernel(const int* __restrict__ src, const int* __restrict__ dst,
                                  const float* __restrict__ ea, const float* __restrict__ ealoop,
                                  const float* __restrict__ ve, const float* __restrict__ asrc,
                                  const float* __restrict__ adst, float* __restrict__ lg,
                                  float* __restrict__ m) {
  int e = blockIdx.x * blockDim.x + threadIdx.x;
  if (e >= EP) return;
  int s, d;
  const float* p;
  if (e < EE) { s = src[e]; d = dst[e]; p = ea + (size_t)e * EDIM; }
  else        { s = d = e - EE;         p = ealoop + (size_t)(e - EE) * EDIM; }
  float ae[HEADS] = {0.f, 0.f, 0.f, 0.f};
#pragma unroll
  for (int k = 0; k < EDIM; k++) {
    float v = p[k];
#pragma unroll
    for (int h = 0; h < HEADS; h++) ae[h] += v * ve[h * EDIM + k];
  }
#pragma unroll
  for (int h = 0; h < HEADS; h++) {
    float lgt = asrc[s * 4 + h] + adst[d * 4 + h] + ae[h];
    lgt = lgt > 0.f ? lgt : NEG_SLOPE * lgt;
    lg[(size_t)e * 4 + h] = lgt;
    atomicMaxF(&m[d * 4 + h], lgt);
  }
}

// edge pass 2: one wave32 per edge — den += ex; out[dst] += ex * h[src] (8 ch per lane)
__global__ void edge_scatter_kernel(const int* __restrict__ src, const int* __restrict__ dst,
                                    const float* __restrict__ lg, const float* __restrict__ m,
                                    float* __restrict__ den, const float* __restrict__ hlin,
                                    float* __restrict__ out) {
  int wid = (blockIdx.x * blockDim.x + threadIdx.x) >> 5;
  int lane = threadIdx.x & 31;
  if (wid >= EP) return;
  int e = wid, s, d;
  if (e < EE) { s = src[e]; d = dst[e]; } else { s = d = e - EE; }
  int h = lane >> 3;                              // lane covers 8 ch inside one head
  float ex = __expf(lg[(size_t)e * 4 + h] - m[d * 4 + h]);
  if ((lane & 7) == 0) atomicAdd(&den[d * 4 + h], ex);
  const float* hp = hlin + (size_t)s * HID + lane * 8;
  float* op = out + (size_t)d * HID + lane * 8;
  float4 v0 = *(const float4*)hp;
  float4 v1 = *(const float4*)(hp + 4);
  atomicAdd(op + 0, ex * v0.x); atomicAdd(op + 1, ex * v0.y);
  atomicAdd(op + 2, ex * v0.z); atomicAdd(op + 3, ex * v0.w);
  atomicAdd(op + 4, ex * v1.x); atomicAdd(op + 5, ex * v1.y);
  atomicAdd(op + 6, ex * v1.z); atomicAdd(op + 7, ex * v1.w);
}

// out = relu(out/den + b), in place
__global__ void finalize_kernel(float* __restrict__ out, const float* __restrict__ den,
                                const float* __restrict__ b) {
  int i = blockIdx.x * blockDim.x + threadIdx.x;
  if (i >= NN * HID) return;
  int n = i >> 8, c = i & 255;
  float v = out[i] / den[n * 4 + (c >> 6)] + b[c];
  out[i] = v > 0.f ? v : 0.f;
}

// ---------------- driver ----------------
static inline int cdiv(long long a, long long b) { return (int)((a + b - 1) / b); }

extern "C" void kernel_launch(void* const* d_in, const int* in_sizes, int n_in,
                              void* d_out, int out_size, void* d_ws, size_t ws_size,
                              hipStream_t stream) {
  const float* x    = (const float*)d_in[0];
  const int*   ei   = (const int*)d_in[1];
  const int*   src  = ei;
  const int*   dst  = ei + EE;
  const float* ea   = (const float*)d_in[2];
  const float* W1   = (const float*)d_in[3];
  const float* We1  = (const float*)d_in[4];
  const float* aS1  = (const float*)d_in[5];
  const float* aD1  = (const float*)d_in[6];
  const float* aE1  = (const float*)d_in[7];
  const float* b1   = (const float*)d_in[8];
  const float* W2   = (const float*)d_in[9];
  const float* We2  = (const float*)d_in[10];
  const float* aS2  = (const float*)d_in[11];
  const float* aD2  = (const float*)d_in[12];
  const float* aE2  = (const float*)d_in[13];
  const float* b2   = (const float*)d_in[14];
  const float* Wo   = (const float*)d_in[15];
  const float* bo   = (const float*)d_in[16];

  float* ws   = (float*)d_ws;
  float* B0   = ws + OFF_B0;
  float* B1   = ws + OFF_B1;
  float* LG   = ws + OFF_LG;
  float* EAL  = ws + OFF_EAL;
  float* DEG  = ws + OFF_DEG;
  float* M    = ws + OFF_M;
  float* DEN  = ws + OFF_DEN;
  float* ASRC = ws + OFF_ASRC;
  float* ADST = ws + OFF_ADST;
  float* VE1  = ws + OFF_VE1;
  float* VE2  = ws + OFF_VE2;

  const int T = 256;
  const int rowTiles = NN / 16;                 // 3125

  // self-loop edge_attr = mean of incoming edge_attr
  fill_kernel<<<cdiv(NN, T), T, 0, stream>>>(DEG, 0.f, NN);
  fill_kernel<<<cdiv((long long)NN * EDIM, T), T, 0, stream>>>(EAL, 0.f, NN * EDIM);
  deg_ea_kernel<<<cdiv(EE, T), T, 0, stream>>>(dst, ea, DEG, EAL);
  ea_div_kernel<<<cdiv((long long)NN * EDIM, T), T, 0, stream>>>(EAL, DEG);

  // fold We*a_edge once per layer
  ve_kernel<<<1, HEADS * EDIM, 0, stream>>>(We1, aE1, VE1);
  ve_kernel<<<1, HEADS * EDIM, 0, stream>>>(We2, aE2, VE2);

  // ---------------- layer 1 ----------------
  {
    int nPanels = HID / 64;
    gemm_wmma_kernel<<<cdiv(rowTiles, 8) * nPanels, T, (size_t)NDIM * 64 * 4, stream>>>(
        x, W1, B0, nullptr, NDIM, HID, rowTiles, nPanels);
  }
  alpha_kernel<<<cdiv((long long)NN * HEADS, T), T, 0, stream>>>(B0, aS1, aD1, ASRC, ADST);
  fill_kernel<<<cdiv((long long)NN * HEADS, T), T, 0, stream>>>(M, -INFINITY, NN * HEADS);
  fill_kernel<<<cdiv((long long)NN * HEADS, T), T, 0, stream>>>(DEN, 0.f, NN * HEADS);
  fill_kernel<<<cdiv((long long)NN * HID, T), T, 0, stream>>>(B1, 0.f, NN * HID);
  edge_logit_kernel<<<cdiv(EP, T), T, 0, stream>>>(src, dst, ea, EAL, VE1, ASRC, ADST, LG, M);
  edge_scatter_kernel<<<cdiv((long long)EP * 32, T), T, 0, stream>>>(src, dst, LG, M, DEN, B0, B1);
  finalize_kernel<<<cdiv((long long)NN * HID, T), T, 0, stream>>>(B1, DEN, b1);

  // ---------------- layer 2 ----------------
  {
    int nPanels = HID / 64;
    gemm_wmma_kernel<<<cdiv(rowTiles, 8) * nPanels, T, (size_t)HID * 64 * 4, stream>>>(
        B1, W2, B0, nullptr, HID, HID, rowTiles, nPanels);
  }
  alpha_kernel<<<cdiv((long long)NN * HEADS, T), T, 0, stream>>>(B0, aS2, aD2, ASRC, ADST);
  fill_kernel<<<cdiv((long long)NN * HEADS, T), T, 0, stream>>>(M, -INFINITY, NN * HEADS);
  fill_kernel<<<cdiv((long long)NN * HEADS, T), T, 0, stream>>>(DEN, 0.f, NN * HEADS);
  fill_kernel<<<cdiv((long long)NN * HID, T), T, 0, stream>>>(B1, 0.f, NN * HID);
  edge_logit_kernel<<<cdiv(EP, T), T, 0, stream>>>(src, dst, ea, EAL, VE2, ASRC, ADST, LG, M);
  edge_scatter_kernel<<<cdiv((long long)EP * 32, T), T, 0, stream>>>(src, dst, LG, M, DEN, B0, B1);
  finalize_kernel<<<cdiv((long long)NN * HID, T), T, 0, stream>>>(B1, DEN, b2);

  // ---------------- output linear (with bias) ----------------
  {
    int nPanels = ODIM / 64;
    gemm_wmma_kernel<<<cdiv(rowTiles, 8) * nPanels, T, (size_t)HID * 64 * 4, stream>>>(
        B1, Wo, (float*)d_out, bo, HID, ODIM, rowTiles, nPanels);
  }
}